// GATLayers_65661460021774
// MI455X (gfx1250) — compile-verified
//
#include <hip/hip_runtime.h>
#include <hip/hip_bf16.h>
#include <math.h>

// ---------------------------------------------------------------------------
// CDNA5 / gfx1250, wave32. GEMMs use V_WMMA_F32_16X16X4_F32 (full fp32),
// software-pipelined (explicit next-step fragments, last iteration peeled) so
// loads for step k+4 are in flight across step k's WMMAs.
// Edge softmax/aggregation is memory/L2-bound; coalesced gathers + f32 atomics.
// ---------------------------------------------------------------------------

typedef float v2f __attribute__((ext_vector_type(2)));
typedef float v8f __attribute__((ext_vector_type(8)));

#define HD 64
#define NEG_ATT 0.2f
#define NEG_ACT 0.01f
#define BN_EPS 1e-5f

// ---------------------------------------------------------------------------
// Generic fills
// ---------------------------------------------------------------------------
__global__ void fill_kernel(float* __restrict__ p, long n, float v) {
    long t = (long)blockIdx.x * blockDim.x + threadIdx.x;
    if (t < n) p[t] = v;
}

__global__ void fill_bias_kernel(float* __restrict__ p, const float* __restrict__ bias,
                                 long total, int C) {
    long t = (long)blockIdx.x * blockDim.x + threadIdx.x;
    if (t < total) p[t] = bias[t % C];
}

// ---------------------------------------------------------------------------
// C[n][o] = sum_k X[n][k] * W[o][k]   (X: NxK row-major, W: OxK row-major)
// Block = 256 threads = 8 waves. Wave w -> rows [ (bx*8+w)*16 , +16 ),
// cols [ by*64 , +64 ) as 4 WMMA 16x16 accumulators.
// A 16x4 fp32 fragment (ISA 7.12.2): lane m=lane&15, kb=2*(lane>>4),
//   a = { X[m][k+kb], X[m][k+kb+1] }.  B 4x16 mirrors A with N in lanes.
// C/D: vgpr v, lane l -> row = v + 8*(l>=16), col = l&15.
// Out-of-range rows handled by CLAMPING the row index: garbage only lands in
// accumulator rows that are never stored; inner loop is branch-free and EXEC
// stays all-ones (WMMA requirement).
// ---------------------------------------------------------------------------
__global__ __launch_bounds__(256)
void gemm_wmma_kernel(const float* __restrict__ X, const float* __restrict__ W,
                      float* __restrict__ C, int N, int K, int O) {
    const int lane = threadIdx.x & 31;
    const int wave = threadIdx.x >> 5;
    const int row0 = (blockIdx.x * 8 + wave) * 16;
    const int col0 = blockIdx.y * 64;
    if (row0 >= N) return;   // wave-uniform exit

    const int m   = lane & 15;
    const int kb  = (lane >> 4) << 1;
    int row = row0 + m;
    if (row > N - 1) row = N - 1;        // clamp: dup rows, never stored

    v8f acc0 = {}, acc1 = {}, acc2 = {}, acc3 = {};

    const float* xp  = X + (size_t)row * K + kb;
    const float* wp0 = W + (size_t)(col0 +  0 + m) * K + kb;
    const float* wp1 = W + (size_t)(col0 + 16 + m) * K + kb;
    const float* wp2 = W + (size_t)(col0 + 32 + m) * K + kb;
    const float* wp3 = W + (size_t)(col0 + 48 + m) * K + kb;

    // prologue: fragments for k = 0
    float2 a_c  = *reinterpret_cast<const float2*>(xp);
    float2 b0_c = *reinterpret_cast<const float2*>(wp0);
    float2 b1_c = *reinterpret_cast<const float2*>(wp1);
    float2 b2_c = *reinterpret_cast<const float2*>(wp2);
    float2 b3_c = *reinterpret_cast<const float2*>(wp3);

    #pragma unroll 4
    for (int k = 4; k < K; k += 4) {
        // issue next step's loads before consuming current fragments
        float2 a_n  = *reinterpret_cast<const float2*>(xp  + k);
        float2 b0_n = *reinterpret_cast<const float2*>(wp0 + k);
        float2 b1_n = *reinterpret_cast<const float2*>(wp1 + k);
        float2 b2_n = *reinterpret_cast<const float2*>(wp2 + k);
        float2 b3_n = *reinterpret_cast<const float2*>(wp3 + k);

        v2f a;  a.x  = a_c.x;  a.y  = a_c.y;
        v2f b0; b0.x = b0_c.x; b0.y = b0_c.y;
        v2f b1; b1.x = b1_c.x; b1.y = b1_c.y;
        v2f b2; b2.x = b2_c.x; b2.y = b2_c.y;
        v2f b3; b3.x = b3_c.x; b3.y = b3_c.y;
        acc0 = __builtin_amdgcn_wmma_f32_16x16x4_f32(false, a, false, b0, (short)0, acc0, false, false);
        acc1 = __builtin_amdgcn_wmma_f32_16x16x4_f32(false, a, false, b1, (short)0, acc1, false, false);
        acc2 = __builtin_amdgcn_wmma_f32_16x16x4_f32(false, a, false, b2, (short)0, acc2, false, false);
        acc3 = __builtin_amdgcn_wmma_f32_16x16x4_f32(false, a, false, b3, (short)0, acc3, false, false);

        a_c = a_n; b0_c = b0_n; b1_c = b1_n; b2_c = b2_n; b3_c = b3_n;
    }

    // epilogue: last step
    {
        v2f a;  a.x  = a_c.x;  a.y  = a_c.y;
        v2f b0; b0.x = b0_c.x; b0.y = b0_c.y;
        v2f b1; b1.x = b1_c.x; b1.y = b1_c.y;
        v2f b2; b2.x = b2_c.x; b2.y = b2_c.y;
        v2f b3; b3.x = b3_c.x; b3.y = b3_c.y;
        acc0 = __builtin_amdgcn_wmma_f32_16x16x4_f32(false, a, false, b0, (short)0, acc0, false, false);
        acc1 = __builtin_amdgcn_wmma_f32_16x16x4_f32(false, a, false, b1, (short)0, acc1, false, false);
        acc2 = __builtin_amdgcn_wmma_f32_16x16x4_f32(false, a, false, b2, (short)0, acc2, false, false);
        acc3 = __builtin_amdgcn_wmma_f32_16x16x4_f32(false, a, false, b3, (short)0, acc3, false, false);
    }

    const int rbase = row0 + ((lane >> 4) << 3);
    const int cn    = lane & 15;
    #pragma unroll
    for (int v = 0; v < 8; ++v) {
        const int r = rbase + v;
        if (r < N) {
            float* cp = C + (size_t)r * O + col0 + cn;
            cp[0]  = acc0[v];
            cp[16] = acc1[v];
            cp[32] = acc2[v];
            cp[48] = acc3[v];
        }
    }
}

// ---------------------------------------------------------------------------
// Per-node attention logits: al_s[n,h] = <h[n,h,:], a_s[h,:]>, same for a_d.
// ---------------------------------------------------------------------------
__global__ void att_logits_kernel(const float* __restrict__ Hf,
                                  const float* __restrict__ a_s,
                                  const float* __restrict__ a_d,
                                  float* __restrict__ al_s,
                                  float* __restrict__ al_d,
                                  int N, int NH) {
    long t = (long)blockIdx.x * blockDim.x + threadIdx.x;
    if (t >= (long)N * NH) return;
    int n = (int)(t / NH), h = (int)(t - (long)n * NH);
    const float4* hp = (const float4*)(Hf + (size_t)n * NH * HD + h * HD);
    const float4* as = (const float4*)(a_s + h * HD);
    const float4* ad = (const float4*)(a_d + h * HD);
    float ss = 0.0f, sd = 0.0f;
    #pragma unroll
    for (int c = 0; c < HD / 4; ++c) {
        float4 v = hp[c], s4 = as[c], d4 = ad[c];
        ss += v.x * s4.x + v.y * s4.y + v.z * s4.z + v.w * s4.w;
        sd += v.x * d4.x + v.y * d4.y + v.z * d4.z + v.w * d4.w;
    }
    al_s[t] = ss;
    al_d[t] = sd;
}

// ---------------------------------------------------------------------------
// float atomic max via int/uint monotone bit trick (init to -FLT_MAX)
// ---------------------------------------------------------------------------
__device__ __forceinline__ void atomicMaxF(float* addr, float v) {
    if (v >= 0.0f) atomicMax((int*)addr, __float_as_int(v));
    else           atomicMin((unsigned int*)addr, __float_as_uint(v));
}

__device__ __forceinline__ void edge_sd(const int* __restrict__ ei, int E, int e,
                                        int& s, int& d) {
    if (e < E) { s = ei[e]; d = ei[E + e]; }
    else       { s = e - E; d = e - E; }     // self loop
}

// Pass 1: per-(edge, head) attention score -> segment max over dst
__global__ void edge_max_kernel(const int* __restrict__ ei, int E, int N,
                                const float* __restrict__ al_s,
                                const float* __restrict__ al_d,
                                float* __restrict__ mmax, int NH) {
    long t = (long)blockIdx.x * blockDim.x + threadIdx.x;
    long total = (long)(E + N) * NH;
    if (t >= total) return;
    int e = (int)(t / NH), h = (int)(t - (long)e * NH);
    int s, d; edge_sd(ei, E, e, s, d);
    float x = al_s[(size_t)s * NH + h] + al_d[(size_t)d * NH + h];
    x = (x > 0.0f) ? x : NEG_ATT * x;        // LeakyReLU(0.2)
    atomicMaxF(&mmax[(size_t)d * NH + h], x);
}

// Pass 2: ex = exp(e - max[dst]); store per-edge, accumulate denominator
__global__ void edge_expsum_kernel(const int* __restrict__ ei, int E, int N,
                                   const float* __restrict__ al_s,
                                   const float* __restrict__ al_d,
                                   const float* __restrict__ mmax,
                                   float* __restrict__ exbuf,
                                   float* __restrict__ den, int NH) {
    long t = (long)blockIdx.x * blockDim.x + threadIdx.x;
    long total = (long)(E + N) * NH;
    if (t >= total) return;
    int e = (int)(t / NH), h = (int)(t - (long)e * NH);
    int s, d; edge_sd(ei, E, e, s, d);
    float x = al_s[(size_t)s * NH + h] + al_d[(size_t)d * NH + h];
    x = (x > 0.0f) ? x : NEG_ATT * x;
    float ex = expf(x - mmax[(size_t)d * NH + h]);
    exbuf[t] = ex;
    atomicAdd(&den[(size_t)d * NH + h], ex);
}

// Pass 3: out[dst] += alpha * h[src].  64 threads per edge (one per channel),
// coalesced gather + coalesced atomic scatter; loop over heads.
__global__ void edge_scatter_kernel(const int* __restrict__ ei, int E, int N,
                                    const float* __restrict__ Hf,
                                    const float* __restrict__ exbuf,
                                    const float* __restrict__ den,
                                    float* __restrict__ out, int NH) {
    long t = (long)blockIdx.x * blockDim.x + threadIdx.x;
    long total = (long)(E + N) * HD;
    if (t >= total) return;
    int e = (int)(t >> 6);
    int c = (int)(t & 63);
    int s, d; edge_sd(ei, E, e, s, d);
    const int C = NH * HD;
    for (int h = 0; h < NH; ++h) {
        float alpha = exbuf[(size_t)e * NH + h] / den[(size_t)d * NH + h];
        float v = Hf[(size_t)s * C + h * HD + c] * alpha;
        atomicAdd(&out[(size_t)d * C + h * HD + c], v);
    }
}

// ---------------------------------------------------------------------------
// BatchNorm (training mode, biased var) over node dim, C = 256 channels.
// ---------------------------------------------------------------------------
__global__ __launch_bounds__(256)
void bn_stats_kernel(const float* __restrict__ agg, const float* __restrict__ bias,
                     float* __restrict__ sum, float* __restrict__ sumsq, int N) {
    const int c  = threadIdx.x;                 // 0..255
    const int r0 = blockIdx.x * 128;
    const int r1 = (r0 + 128 < N) ? (r0 + 128) : N;
    const float b = bias[c];
    float s = 0.0f, sq = 0.0f;
    for (int r = r0; r < r1; ++r) {
        float v = agg[(size_t)r * 256 + c] + b;
        s += v; sq += v * v;
    }
    atomicAdd(&sum[c], s);
    atomicAdd(&sumsq[c], sq);
}

__global__ void bn_apply_kernel(const float* __restrict__ agg,
                                const float* __restrict__ bias,
                                const float* __restrict__ sum,
                                const float* __restrict__ sumsq,
                                const float* __restrict__ gamma,
                                const float* __restrict__ beta,
                                const float* __restrict__ residual,
                                float* __restrict__ out, int N) {
    long t = (long)blockIdx.x * blockDim.x + threadIdx.x;
    if (t >= (long)N * 256) return;
    int c = (int)(t & 255);
    float invN = 1.0f / (float)N;
    float mu  = sum[c] * invN;
    float var = sumsq[c] * invN - mu * mu;
    float v = agg[t] + bias[c];
    float y = gamma[c] * (v - mu) * rsqrtf(var + BN_EPS) + beta[c];
    y = (y > 0.0f) ? y : NEG_ACT * y;           // LeakyReLU(0.01)
    if (residual) y += residual[t];
    out[t] = y;
}

// ---------------------------------------------------------------------------
// Host orchestration
// ---------------------------------------------------------------------------
static inline int cdiv(long a, long b) { return (int)((a + b - 1) / b); }

extern "C" void kernel_launch(void* const* d_in, const int* in_sizes, int n_in,
                              void* d_out, int out_size, void* d_ws, size_t ws_size,
                              hipStream_t stream) {
    const float* x     = (const float*)d_in[0];
    const int*   ei    = (const int*)  d_in[1];
    const float* W1    = (const float*)d_in[2];
    const float* a1s   = (const float*)d_in[3];
    const float* a1d   = (const float*)d_in[4];
    const float* b1    = (const float*)d_in[5];
    const float* W2    = (const float*)d_in[6];
    const float* a2s   = (const float*)d_in[7];
    const float* a2d   = (const float*)d_in[8];
    const float* b2    = (const float*)d_in[9];
    const float* W3    = (const float*)d_in[10];
    const float* a3s   = (const float*)d_in[11];
    const float* a3d   = (const float*)d_in[12];
    const float* b3    = (const float*)d_in[13];
    const float* gamma = (const float*)d_in[14];
    const float* beta  = (const float*)d_in[15];

    const int N = in_sizes[0] / 128;      // x: [N, 2*HD]
    const int E = in_sizes[1] / 2;        // edge_index: [2, E]
    const int Etot = E + N;               // with self loops

    // Workspace carve-up (floats)
    float* w = (float*)d_ws;
    size_t off = 0;
    auto alloc = [&](size_t n) { float* p = w + off; off += n; return p; };
    float* bufA = alloc((size_t)N * 256);   // h1 / h2 / post2
    float* bufB = alloc((size_t)N * 256);   // agg1 / agg2 / h3
    float* bufC = alloc((size_t)N * 256);   // post1 (residual)
    float* als  = alloc((size_t)N * 4);
    float* ald  = alloc((size_t)N * 4);
    float* mbuf = alloc((size_t)N * 4);
    float* den  = alloc((size_t)N * 4);
    float* exb  = alloc((size_t)Etot * 4);
    float* sumb = alloc(256);
    float* sqb  = alloc(256);
    (void)ws_size; (void)n_in; (void)out_size;

    const int T = 256;
    dim3 blk(T);

    auto fill = [&](float* p, long n, float v) {
        fill_kernel<<<cdiv(n, T), blk, 0, stream>>>(p, n, v);
    };

    auto gat_layer = [&](const float* hin, const float* Wm, const float* asv,
                         const float* adv, int K, int NH, float* hbuf, float* aggbuf) {
        const int O = NH * HD;
        dim3 ggrid(cdiv(N, 128), O / 64);
        gemm_wmma_kernel<<<ggrid, blk, 0, stream>>>(hin, Wm, hbuf, N, K, O);
        att_logits_kernel<<<cdiv((long)N * NH, T), blk, 0, stream>>>(
            hbuf, asv, adv, als, ald, N, NH);
        fill(mbuf, (long)N * NH, -3.402823466e38f);
        fill(den,  (long)N * NH, 0.0f);
        fill(aggbuf, (long)N * O, 0.0f);
        edge_max_kernel<<<cdiv((long)Etot * NH, T), blk, 0, stream>>>(
            ei, E, N, als, ald, mbuf, NH);
        edge_expsum_kernel<<<cdiv((long)Etot * NH, T), blk, 0, stream>>>(
            ei, E, N, als, ald, mbuf, exb, den, NH);
        edge_scatter_kernel<<<cdiv((long)Etot * HD, T), blk, 0, stream>>>(
            ei, E, N, hbuf, exb, den, aggbuf, NH);
    };

    auto bn_lrelu = [&](const float* agg, const float* bias, const float* residual,
                        float* outp) {
        fill(sumb, 256, 0.0f);
        fill(sqb, 256, 0.0f);
        bn_stats_kernel<<<cdiv(N, 128), blk, 0, stream>>>(agg, bias, sumb, sqb, N);
        bn_apply_kernel<<<cdiv((long)N * 256, T), blk, 0, stream>>>(
            agg, bias, sumb, sqb, gamma, beta, residual, outp, N);
    };

    // ---- Layer 1: GAT(128 -> 4x64, concat) + bias + BN + LeakyReLU ----
    gat_layer(x, W1, a1s, a1d, 2 * HD, 4, bufA, bufB);
    bn_lrelu(bufB, b1, nullptr, bufC);                 // post1 (also residual)

    // ---- Layer 2: GAT(256 -> 4x64, concat) + bias + BN + LeakyReLU + res ----
    gat_layer(bufC, W2, a2s, a2d, 4 * HD, 4, bufA, bufB);
    bn_lrelu(bufB, b2, bufC, bufA);                    // post2 -> bufA

    // ---- Layer 3: GAT(256 -> 1x64) + bias, straight into d_out ----
    float* outp = (float*)d_out;
    {
        const int NH = 1, K = 4 * HD, O = HD;
        dim3 ggrid(cdiv(N, 128), O / 64);
        gemm_wmma_kernel<<<ggrid, blk, 0, stream>>>(bufA, W3, bufB, N, K, O);
        att_logits_kernel<<<cdiv((long)N * NH, T), blk, 0, stream>>>(
            bufB, a3s, a3d, als, ald, N, NH);
        fill(mbuf, (long)N * NH, -3.402823466e38f);
        fill(den,  (long)N * NH, 0.0f);
        // init output with bias (b3), scatter accumulates on top
        fill_bias_kernel<<<cdiv((long)N * O, T), blk, 0, stream>>>(
            outp, b3, (long)N * O, O);
        edge_max_kernel<<<cdiv((long)Etot * NH, T), blk, 0, stream>>>(
            ei, E, N, als, ald, mbuf, NH);
        edge_expsum_kernel<<<cdiv((long)Etot * NH, T), blk, 0, stream>>>(
            ei, E, N, als, ald, mbuf, exb, den, NH);
        edge_scatter_kernel<<<cdiv((long)Etot * HD, T), blk, 0, stream>>>(
            ei, E, N, bufB, exb, den, outp, NH);
    }
}